// SSIM_5798205850258
// MI455X (gfx1250) — compile-verified
//
#include <hip/hip_runtime.h>
#include <hip/hip_bf16.h>

typedef __attribute__((ext_vector_type(2))) float v2f;
typedef __attribute__((ext_vector_type(8))) float v8f;

#define IMG_H 1024
#define IMG_W 1024
#define OUT_H 1016
#define OUT_W 1016
#define NIMG  32
#define TILES_X 64   // ceil(1016/16); 64*16 = 1024 also tiles the L1 domain
#define TILES_Y 64
#define WAVES_PER_BLOCK 8
#define PADW 24      // LDS row pitch in floats (keeps ds_load_b64 8B-aligned)

// ---------------------------------------------------------------------------
// ws layout (floats): [0..255] SSIM partial sums, [256..511] L1 partial sums
// ---------------------------------------------------------------------------

__global__ __launch_bounds__(512) void init_partials(float* p) {
    if (threadIdx.x < 512) p[threadIdx.x] = 0.0f;
}

// ---------------------------------------------------------------------------
// Fused SSIM + L1 kernel: one wave computes one 16x16 output tile.
// BOTH separable 9-tap Gaussian passes run on V_WMMA_F32_16X16X4_F32:
//   pass 1 (vertical):   V(16x32) = Wv(16x24) x T(24x32)   (2 column blocks)
//   pass 2 (horizontal): O(16x16) = V(16x24)  x Wh(24x16)
// Wv in A-layout and Wh in B-layout share identical per-lane registers:
//   val = g[K - (lane&15)], K = 4q + 2*(lane>=16) + {0,1}.
// Inputs are loaded DIRECTLY in WMMA B-layout (lane (h,n) loads rows
// 4q+2h+{0,1} of its column) so no cross-lane shuffles are needed, and the
// q=0..3 left-block registers are exactly the tile's owned 16x16 pixels,
// which also feeds the L1 |X-Y| term with zero extra memory traffic.
// ---------------------------------------------------------------------------
__global__ __launch_bounds__(256) void ssim_main(const float* __restrict__ X,
                                                 const float* __restrict__ Y,
                                                 const float* __restrict__ w2d,
                                                 float* __restrict__ partial) {
    // per-wave staging of V (5 maps x 16 rows x 24 cols), plus padded g table
    __shared__ float lds[WAVES_PER_BLOCK][5][16][PADW];
    __shared__ float gtab[48];   // gtab[16+d] = g[d] for d in [0,8], else 0

    const int tid  = threadIdx.x;
    const int lane = tid & 31;
    const int wave = tid >> 5;
    const int half = lane >> 4;     // 0: lanes 0-15, 1: lanes 16-31
    const int nIdx = lane & 15;     // A-row (M) / B-col (N) index

    // --- zero-padded 1-D gaussian table (g from w2d = outer(g,g)) ----------
    if (tid < 48) {
        float v = 0.0f;
        if (tid >= 16 && tid <= 24)
            v = w2d[(tid - 16) * 9 + 4] * rsqrtf(w2d[4 * 9 + 4]);
        gtab[tid] = v;
    }
    __syncthreads();

    // tile assignment
    const int tilesPerImg = TILES_X * TILES_Y;
    int tileId = blockIdx.x * WAVES_PER_BLOCK + wave;
    int img = tileId / tilesPerImg;
    int t   = tileId % tilesPerImg;
    int ty = t / TILES_X, tx = t % TILES_X;
    int row0 = ty * 16, col0 = tx * 16;

    // banded Toeplitz weight chunks (shared by pass-1 A and pass-2 B)
    v2f bw[6];
    #pragma unroll
    for (int q = 0; q < 6; ++q) {
        int K0 = 4 * q + 2 * half;
        bw[q].x = gtab[16 + K0 - nIdx];
        bw[q].y = gtab[16 + K0 + 1 - nIdx];
    }

    // --- load input slab directly in WMMA B-layout -------------------------
    // lane (h, n): rows 4q+2h+{0,1}, columns col0+n (L) and col0+16+n (R)
    const float* xp = X + (size_t)img * (IMG_H * (size_t)IMG_W);
    const float* yp = Y + (size_t)img * (IMG_H * (size_t)IMG_W);
    int cL = col0 + nIdx;      if (cL > IMG_W - 1) cL = IMG_W - 1;
    int cR = col0 + 16 + nIdx; if (cR > IMG_W - 1) cR = IMG_W - 1;
    v2f xL[6], yL[6], xR[6], yR[6];
    #pragma unroll
    for (int q = 0; q < 6; ++q) {
        int r0 = row0 + 4 * q + 2 * half;
        int ra = r0;     if (ra > IMG_H - 1) ra = IMG_H - 1;  // edge clamp
        int rb = r0 + 1; if (rb > IMG_H - 1) rb = IMG_H - 1;
        size_t oa = (size_t)ra * IMG_W, ob = (size_t)rb * IMG_W;
        xL[q].x = xp[oa + cL]; xL[q].y = xp[ob + cL];
        yL[q].x = yp[oa + cL]; yL[q].y = yp[ob + cL];
        xR[q].x = xp[oa + cR]; xR[q].y = xp[ob + cR];
        yR[q].x = yp[oa + cR]; yR[q].y = yp[ob + cR];
    }

    // --- L1 term over this tile's owned 16x16 pixels (q=0..3, left block) --
    // rows 4q+2h+{0,1} <= 15 and col cL are never clamped for owned pixels.
    float lsum = 0.0f;
    #pragma unroll
    for (int q = 0; q < 4; ++q) {
        lsum += fabsf(xL[q].x - yL[q].x) + fabsf(xL[q].y - yL[q].y);
    }

    // --- pass 1: vertical conv on WMMA for 5 maps, two 16-col blocks -------
    // product maps are componentwise products of the x/y B operands.
    const v8f vzero = {0,0,0,0,0,0,0,0};
    v8f accL[5], accR[5];
    #pragma unroll
    for (int m = 0; m < 5; ++m) { accL[m] = vzero; accR[m] = vzero; }

    #pragma unroll
    for (int q = 0; q < 6; ++q) {
        v2f BxL = xL[q], ByL = yL[q], BxR = xR[q], ByR = yR[q];
        v2f BxxL = BxL * BxL, ByyL = ByL * ByL, BxyL = BxL * ByL;
        v2f BxxR = BxR * BxR, ByyR = ByR * ByR, BxyR = BxR * ByR;

        accL[0] = __builtin_amdgcn_wmma_f32_16x16x4_f32(false, bw[q], false, BxL,  (short)0, accL[0], false, false);
        accL[1] = __builtin_amdgcn_wmma_f32_16x16x4_f32(false, bw[q], false, ByL,  (short)0, accL[1], false, false);
        accL[2] = __builtin_amdgcn_wmma_f32_16x16x4_f32(false, bw[q], false, BxxL, (short)0, accL[2], false, false);
        accL[3] = __builtin_amdgcn_wmma_f32_16x16x4_f32(false, bw[q], false, ByyL, (short)0, accL[3], false, false);
        accL[4] = __builtin_amdgcn_wmma_f32_16x16x4_f32(false, bw[q], false, BxyL, (short)0, accL[4], false, false);
        accR[0] = __builtin_amdgcn_wmma_f32_16x16x4_f32(false, bw[q], false, BxR,  (short)0, accR[0], false, false);
        accR[1] = __builtin_amdgcn_wmma_f32_16x16x4_f32(false, bw[q], false, ByR,  (short)0, accR[1], false, false);
        accR[2] = __builtin_amdgcn_wmma_f32_16x16x4_f32(false, bw[q], false, BxxR, (short)0, accR[2], false, false);
        accR[3] = __builtin_amdgcn_wmma_f32_16x16x4_f32(false, bw[q], false, ByyR, (short)0, accR[3], false, false);
        accR[4] = __builtin_amdgcn_wmma_f32_16x16x4_f32(false, bw[q], false, BxyR, (short)0, accR[4], false, false);
    }

    // --- V (C/D layout) -> LDS row-major; only cols 0..23 are needed -------
    #pragma unroll
    for (int m = 0; m < 5; ++m) {
        #pragma unroll
        for (int e = 0; e < 8; ++e) {
            int r = e + 8 * half;
            lds[wave][m][r][nIdx] = accL[m][e];
            if (nIdx < 8) lds[wave][m][r][16 + nIdx] = accR[m][e];
        }
    }
    // Per-wave LDS; DS ops are in-order within a wave -> no barrier needed.

    // --- pass 2: horizontal conv O = V(16x24) x Wh(24x16) ------------------
    v8f a0 = vzero, a1 = vzero, a2 = vzero, a3 = vzero, a4 = vzero;
    #pragma unroll
    for (int q = 0; q < 6; ++q) {
        int cb = 4 * q + 2 * half;
        v2f A0 = *(const v2f*)&lds[wave][0][nIdx][cb];
        v2f A1 = *(const v2f*)&lds[wave][1][nIdx][cb];
        v2f A2 = *(const v2f*)&lds[wave][2][nIdx][cb];
        v2f A3 = *(const v2f*)&lds[wave][3][nIdx][cb];
        v2f A4 = *(const v2f*)&lds[wave][4][nIdx][cb];
        a0 = __builtin_amdgcn_wmma_f32_16x16x4_f32(false, A0, false, bw[q], (short)0, a0, false, false);
        a1 = __builtin_amdgcn_wmma_f32_16x16x4_f32(false, A1, false, bw[q], (short)0, a1, false, false);
        a2 = __builtin_amdgcn_wmma_f32_16x16x4_f32(false, A2, false, bw[q], (short)0, a2, false, false);
        a3 = __builtin_amdgcn_wmma_f32_16x16x4_f32(false, A3, false, bw[q], (short)0, a3, false, false);
        a4 = __builtin_amdgcn_wmma_f32_16x16x4_f32(false, A4, false, bw[q], (short)0, a4, false, false);
    }

    // --- SSIM map directly in C/D layout: element e -> (e + 8*half, nIdx) --
    const float C1 = 0.0004f;   // (0.01*2)^2
    const float C2 = 0.0036f;   // (0.03*2)^2
    float wsum = 0.0f;
    #pragma unroll
    for (int e = 0; e < 8; ++e) {
        float ux = a0[e], uy = a1[e], uxx = a2[e], uyy = a3[e], uxy = a4[e];
        float vx  = uxx - ux * ux;
        float vy  = uyy - uy * uy;
        float vxy = uxy - ux * uy;
        float A1v = 2.0f * ux * uy + C1;
        float A2v = 2.0f * vxy + C2;
        float B1v = ux * ux + uy * uy + C1;
        float B2v = vx + vy + C2;
        float S = (A1v * A2v) / (B1v * B2v);
        int r = row0 + e + 8 * half;
        int c = col0 + nIdx;
        if (r < OUT_H && c < OUT_W) wsum += 1.0f - S;
    }

    // wave32 reduce both sums, then scattered fp32 atomics
    #pragma unroll
    for (int off = 16; off > 0; off >>= 1) {
        wsum += __shfl_xor(wsum, off, 32);
        lsum += __shfl_xor(lsum, off, 32);
    }
    if (lane == 0) {
        int slot = (blockIdx.x * WAVES_PER_BLOCK + wave) & 255;
        __hip_atomic_fetch_add(&partial[slot], wsum,
                               __ATOMIC_RELAXED, __HIP_MEMORY_SCOPE_AGENT);
        __hip_atomic_fetch_add(&partial[256 + slot], lsum,
                               __ATOMIC_RELAXED, __HIP_MEMORY_SCOPE_AGENT);
    }
}

// ---------------------------------------------------------------------------
// Combine: out = 0.5 * mean(1 - S) + 1.0 * mean(|X - Y|)
// ---------------------------------------------------------------------------
__global__ __launch_bounds__(256) void final_kernel(const float* __restrict__ partial,
                                                    float* __restrict__ out) {
    __shared__ float sa[8], sb[8];
    int tid = threadIdx.x, lane = tid & 31, wave = tid >> 5;
    float a = partial[tid];        // SSIM partials
    float b = partial[256 + tid];  // L1 partials
    #pragma unroll
    for (int off = 16; off > 0; off >>= 1) {
        a += __shfl_xor(a, off, 32);
        b += __shfl_xor(b, off, 32);
    }
    if (lane == 0) { sa[wave] = a; sb[wave] = b; }
    __syncthreads();
    if (tid == 0) {
        float A = 0.0f, B = 0.0f;
        #pragma unroll
        for (int i = 0; i < 8; ++i) { A += sa[i]; B += sb[i]; }
        const float invSsim = 1.0f / (32.0f * 1016.0f * 1016.0f);
        const float invL1   = 1.0f / (32.0f * 1024.0f * 1024.0f);
        out[0] = 0.5f * (A * invSsim) + 1.0f * (B * invL1);
    }
}

// ---------------------------------------------------------------------------
extern "C" void kernel_launch(void* const* d_in, const int* in_sizes, int n_in,
                              void* d_out, int out_size, void* d_ws, size_t ws_size,
                              hipStream_t stream) {
    const float* X = (const float*)d_in[0];
    const float* Y = (const float*)d_in[1];
    const float* w = (const float*)d_in[2];
    float* partial = (float*)d_ws;
    float* out = (float*)d_out;

    init_partials<<<1, 512, 0, stream>>>(partial);

    // 32 images * 64*64 tiles / 8 waves per block = 16384 blocks
    const int nBlocks = (NIMG * TILES_X * TILES_Y) / WAVES_PER_BLOCK;
    ssim_main<<<nBlocks, 256, 0, stream>>>(X, Y, w, partial);

    final_kernel<<<1, 256, 0, stream>>>(partial, out);
}